// NSDynamicsNetwork_19155554140916
// MI455X (gfx1250) — compile-verified
//
#include <hip/hip_runtime.h>

// ---------------------------------------------------------------------------
// NS solver: van-Leer advection + viscosity + spectral (DFT-GEMM) projection.
// Poisson solve cast as 12 dense 256x256x256 FP32 GEMMs per projection,
// executed with V_WMMA_F32_16X16X4_F32 (CDNA5 wave32 WMMA).
// GEMM is register-blocked: each wave owns a 16x64 strip (4 accumulators),
// reusing its A fragment across 4 B tiles to cut LDS traffic per WMMA.
// ---------------------------------------------------------------------------

#define NDIM        256
#define N2          (NDIM * NDIM)          // 65536
#define BATCH       8
#define TSTEPS      8                      // num_steps_forward (fixed by setup)
#define INNER_STEPS 10

static constexpr float TWO_PI  = 6.28318530717958647692f;
static constexpr float DXF     = TWO_PI / 256.0f;
static constexpr float INV_DXF = 256.0f / TWO_PI;
static constexpr float VISCF   = 1e-3f;

typedef __attribute__((ext_vector_type(2))) float v2f;
typedef __attribute__((ext_vector_type(8))) float v8f;

__device__ __forceinline__ int pid(int i, int j) {
  return (((i) & 255) << 8) | ((j) & 255);
}

// ---------------------------------------------------------------------------
// DFT factor matrices C[k,x]=cos(2pi kx/N), S[k,x]=sin(2pi kx/N) (symmetric),
// and the pseudo-inverse eigenvalues of the periodic FD Laplacian.
// ---------------------------------------------------------------------------
__global__ void init_dft_mats(float* __restrict__ Cm, float* __restrict__ Sm,
                              float* __restrict__ Lm) {
  int g = blockIdx.x * blockDim.x + threadIdx.x;
  if (g >= N2) return;
  int i = g >> 8, j = g & 255;
  int prod = (i * j) & 255;                 // exact mod-256 phase
  float th = TWO_PI * (float)prod * (1.0f / 256.0f);
  Cm[g] = cosf(th);
  Sm[g] = sinf(th);
  float l1 = (2.0f * cosf(TWO_PI * (float)i * (1.0f / 256.0f)) - 2.0f) * (INV_DXF * INV_DXF);
  float l2 = (2.0f * cosf(TWO_PI * (float)j * (1.0f / 256.0f)) - 2.0f) * (INV_DXF * INV_DXF);
  Lm[g] = (g == 0) ? 0.0f : (1.0f / (l1 + l2));
}

// [B][H][W][3] -> three [B][H][W] planes
__global__ void unpack_kernel(const float* __restrict__ y0,
                              float* __restrict__ u, float* __restrict__ v,
                              float* __restrict__ d) {
  int g = blockIdx.x * blockDim.x + threadIdx.x;   // b*N2 + p
  size_t s = (size_t)g * 3;
  u[g] = y0[s + 0];
  v[g] = y0[s + 1];
  d[g] = y0[s + 2];
}

__global__ void pack_kernel(const float* __restrict__ u, const float* __restrict__ v,
                            const float* __restrict__ d, float* __restrict__ out, int t) {
  int g = blockIdx.x * blockDim.x + threadIdx.x;   // b*N2 + p
  int b = g >> 16;
  int p = g & (N2 - 1);
  size_t o = ((size_t)b * TSTEPS + t) * (size_t)(N2 * 3) + (size_t)p * 3;
  out[o + 0] = u[g];
  out[o + 1] = v[g];
  out[o + 2] = d[g];
}

// ---------------------------------------------------------------------------
// van-Leer flux-limited face flux (TVD blend of upwind and Lax-Wendroff)
// ---------------------------------------------------------------------------
__device__ __forceinline__ float vl_face(float cm1, float c0, float c1, float c2,
                                         float uf, float dtdx) {
  float dc      = c1 - c0;
  float f_low   = (uf >= 0.0f) ? uf * c0 : uf * c1;
  float f_high  = uf * 0.5f * (c0 + c1) - 0.5f * uf * uf * dtdx * dc;
  float dc_up   = (uf >= 0.0f) ? (c0 - cm1) : (c2 - c1);
  float dc_safe = (fabsf(dc) > 1e-12f) ? dc : 1e-12f;
  float r       = dc_up / dc_safe;
  float phi     = (r > 0.0f) ? (2.0f * r / (1.0f + r)) : 0.0f;
  return f_low + phi * (f_high - f_low);
}

__device__ float advect_tend(const float* __restrict__ c, const float* __restrict__ u,
                             const float* __restrict__ v, int i, int j, float dtdx) {
  float t = 0.0f;
  {   // axis -2 (rows), velocity u
    float ufp = 0.5f * (u[pid(i, j)] + u[pid(i + 1, j)]);
    float Fp  = vl_face(c[pid(i - 1, j)], c[pid(i, j)], c[pid(i + 1, j)], c[pid(i + 2, j)], ufp, dtdx);
    float ufm = 0.5f * (u[pid(i - 1, j)] + u[pid(i, j)]);
    float Fm  = vl_face(c[pid(i - 2, j)], c[pid(i - 1, j)], c[pid(i, j)], c[pid(i + 1, j)], ufm, dtdx);
    t -= (Fp - Fm) * INV_DXF;
  }
  {   // axis -1 (cols), velocity v
    float vfp = 0.5f * (v[pid(i, j)] + v[pid(i, j + 1)]);
    float Fp  = vl_face(c[pid(i, j - 1)], c[pid(i, j)], c[pid(i, j + 1)], c[pid(i, j + 2)], vfp, dtdx);
    float vfm = 0.5f * (v[pid(i, j - 1)] + v[pid(i, j)]);
    float Fm  = vl_face(c[pid(i, j - 2)], c[pid(i, j - 1)], c[pid(i, j)], c[pid(i, j + 1)], vfm, dtdx);
    t -= (Fp - Fm) * INV_DXF;
  }
  return t;
}

// u1 = u + dt*(adv(u;u,v) + visc*lap(u)); v1 likewise
__global__ void stencil_uv_kernel(const float* __restrict__ u, const float* __restrict__ v,
                                  float* __restrict__ nu, float* __restrict__ nv,
                                  const float* __restrict__ dtp) {
  int g = blockIdx.x * blockDim.x + threadIdx.x;
  int b = g >> 16;
  int p = g & (N2 - 1);
  int i = p >> 8, j = p & 255;
  const float* ub = u + (size_t)b * N2;
  const float* vb = v + (size_t)b * N2;
  float dt   = dtp[0];
  float dtdx = dt * INV_DXF;
  float lu = (ub[pid(i, j - 1)] + ub[pid(i, j + 1)] + ub[pid(i - 1, j)] + ub[pid(i + 1, j)]
              - 4.0f * ub[p]) * (INV_DXF * INV_DXF);
  float lv = (vb[pid(i, j - 1)] + vb[pid(i, j + 1)] + vb[pid(i - 1, j)] + vb[pid(i + 1, j)]
              - 4.0f * vb[p]) * (INV_DXF * INV_DXF);
  float au = advect_tend(ub, ub, vb, i, j, dtdx);
  float av = advect_tend(vb, ub, vb, i, j, dtdx);
  nu[g] = ub[p] + dt * (au + VISCF * lu);
  nv[g] = vb[p] + dt * (av + VISCF * lv);
}

// density advected with OLD velocity
__global__ void stencil_d_kernel(const float* __restrict__ d, const float* __restrict__ u,
                                 const float* __restrict__ v, float* __restrict__ nd,
                                 const float* __restrict__ dtp) {
  int g = blockIdx.x * blockDim.x + threadIdx.x;
  int b = g >> 16;
  int p = g & (N2 - 1);
  int i = p >> 8, j = p & 255;
  const float* db = d + (size_t)b * N2;
  const float* ub = u + (size_t)b * N2;
  const float* vb = v + (size_t)b * N2;
  float dt = dtp[0];
  nd[g] = db[p] + dt * advect_tend(db, ub, vb, i, j, dt * INV_DXF);
}

// backward-difference divergence
__global__ void divergence_kernel(const float* __restrict__ u, const float* __restrict__ v,
                                  float* __restrict__ dv) {
  int g = blockIdx.x * blockDim.x + threadIdx.x;
  int b = g >> 16;
  int p = g & (N2 - 1);
  int i = p >> 8, j = p & 255;
  const float* ub = u + (size_t)b * N2;
  const float* vb = v + (size_t)b * N2;
  dv[g] = (ub[p] - ub[pid(i - 1, j)]) * INV_DXF + (vb[p] - vb[pid(i, j - 1)]) * INV_DXF;
}

// Rh holds Re(Dhat); Ih holds -(Im(Dhat)); scale by +L / -L respectively.
__global__ void scale_eigs_kernel(float* __restrict__ Rh, float* __restrict__ Ih,
                                  const float* __restrict__ L) {
  int g = blockIdx.x * blockDim.x + threadIdx.x;
  float l = L[g & (N2 - 1)];
  Rh[g] *= l;
  Ih[g] *= -l;
}

// forward-difference gradient subtract
__global__ void apply_proj_kernel(float* __restrict__ u, float* __restrict__ v,
                                  const float* __restrict__ q) {
  int g = blockIdx.x * blockDim.x + threadIdx.x;
  int b = g >> 16;
  int p = g & (N2 - 1);
  int i = p >> 8, j = p & 255;
  const float* qb = q + (size_t)b * N2;
  u[g] -= (qb[pid(i + 1, j)] - qb[p]) * INV_DXF;
  v[g] -= (qb[pid(i, j + 1)] - qb[p]) * INV_DXF;
}

// ---------------------------------------------------------------------------
// FP32 WMMA GEMM: OUT(256x256) = alpha * A(256x256) * B(256x256) + beta * OUT
// Batched via blockIdx.z with element strides (0 => broadcast matrix).
// Block = 128 threads = 4 waves stacked in M -> 64x64 OUT tile.
// Each wave register-blocks a 16x64 strip: one A fragment per K-step feeds
// 4 WMMAs (4 accumulators, 32 VGPRs), so LDS traffic drops to 2.5 dw/WMMA.
// K staged through LDS in 32-wide chunks; inner op: V_WMMA_F32_16X16X4_F32.
// A frag (16x4): lanes 0-15 K={0,1}, lanes 16-31 K={2,3} (ISA 7.12.2).
// D (16x16 f32): VGPR r -> M=r (lanes<16) / M=r+8 (lanes>=16), N=lane&15.
// ---------------------------------------------------------------------------
__global__ void __launch_bounds__(128)
wmma_gemm256(const float* __restrict__ A, long long strideA,
             const float* __restrict__ B, long long strideB,
             float* __restrict__ OUT, long long strideO,
             float alpha, float beta) {
  __shared__ float As[64][33];   // 64 M-rows x 32 K (padded)
  __shared__ float Bs[32][65];   // 32 K-rows x 64 N (padded)
  const int z = blockIdx.z;
  A   += (size_t)z * (size_t)strideA;
  B   += (size_t)z * (size_t)strideB;
  OUT += (size_t)z * (size_t)strideO;

  const int tid  = threadIdx.x;
  const int lane = tid & 31;            // wave32
  const int wave = tid >> 5;            // 0..3 -> M sub-tile
  const int rowBase = blockIdx.y << 6;  // 64-row block
  const int colBase = blockIdx.x << 6;  // 64-col block
  const int l15  = lane & 15;
  const int half = lane >> 4;           // lane half selects K pair {0,1} vs {2,3}
  const int arow = (wave << 4) + l15;

  v8f acc0 = {}, acc1 = {}, acc2 = {}, acc3 = {};

  for (int kc = 0; kc < 256; kc += 32) {
    // A: 64x32, B: 32x64 -> 2048 elems each, 16 per thread, coalesced rows.
    for (int s = tid; s < 2048; s += 128) {
      int ra = s >> 5, ca = s & 31;
      As[ra][ca] = A[(rowBase + ra) * 256 + kc + ca];
      int rb = s >> 6, cb = s & 63;
      Bs[rb][cb] = B[(kc + rb) * 256 + colBase + cb];
    }
    if (kc + 32 < 256) {                // hint next K chunk toward the caches
      __builtin_prefetch(&A[(rowBase + (tid >> 1)) * 256 + kc + 32], 0, 0);
      __builtin_prefetch(&B[(kc + 32 + (tid >> 2)) * 256 + colBase], 0, 0);
    }
    __syncthreads();
#if __has_builtin(__builtin_amdgcn_wmma_f32_16x16x4_f32)
#pragma unroll
    for (int kk = 0; kk < 32; kk += 4) {
      v2f af;
      af.x = As[arow][kk + 2 * half + 0];
      af.y = As[arow][kk + 2 * half + 1];
      const float* br0 = &Bs[kk + 2 * half + 0][0];
      const float* br1 = &Bs[kk + 2 * half + 1][0];
      v2f b0, b1, b2, b3;
      b0.x = br0[ 0 + l15];  b0.y = br1[ 0 + l15];
      b1.x = br0[16 + l15];  b1.y = br1[16 + l15];
      b2.x = br0[32 + l15];  b2.y = br1[32 + l15];
      b3.x = br0[48 + l15];  b3.y = br1[48 + l15];
      acc0 = __builtin_amdgcn_wmma_f32_16x16x4_f32(false, af, false, b0, (short)0, acc0, false, false);
      acc1 = __builtin_amdgcn_wmma_f32_16x16x4_f32(false, af, false, b1, (short)0, acc1, false, false);
      acc2 = __builtin_amdgcn_wmma_f32_16x16x4_f32(false, af, false, b2, (short)0, acc2, false, false);
      acc3 = __builtin_amdgcn_wmma_f32_16x16x4_f32(false, af, false, b3, (short)0, acc3, false, false);
    }
#else
    // Fallback (wmma=0 in the histogram signals the builtin is absent)
#pragma unroll 4
    for (int kk = 0; kk < 32; ++kk) {
      float a = As[arow][kk];
      float w0 = Bs[kk][ 0 + l15], w1 = Bs[kk][16 + l15];
      float w2 = Bs[kk][32 + l15], w3 = Bs[kk][48 + l15];
#pragma unroll
      for (int r = 0; r < 8; ++r) {
        acc0[r] += a * w0; acc1[r] += a * w1;
        acc2[r] += a * w2; acc3[r] += a * w3;
      }
    }
#endif
    __syncthreads();
  }

  const int mtop = rowBase + (wave << 4) + (half << 3);
#pragma unroll
  for (int tcol = 0; tcol < 4; ++tcol) {
    v8f a = (tcol == 0) ? acc0 : (tcol == 1) ? acc1 : (tcol == 2) ? acc2 : acc3;
    const int n = colBase + (tcol << 4) + l15;
#pragma unroll
    for (int r = 0; r < 8; ++r) {
      int m = mtop + r;
      float res = alpha * a[r];
      if (beta != 0.0f) res += beta * OUT[m * 256 + n];
      OUT[m * 256 + n] = res;
    }
  }
}

// ---------------------------------------------------------------------------
// Host orchestration
// ---------------------------------------------------------------------------
extern "C" void kernel_launch(void* const* d_in, const int* in_sizes, int n_in,
                              void* d_out, int out_size, void* d_ws, size_t ws_size,
                              hipStream_t stream) {
  (void)in_sizes; (void)n_in; (void)out_size;
  const float* y0  = (const float*)d_in[0];
  const float* dtp = (const float*)d_in[1];
  float* out = (float*)d_out;

  float* W = (float*)d_ws;
  const size_t msz = (size_t)N2;
  const size_t fld = (size_t)BATCH * N2;
  float* Cm = W;
  float* Sm = Cm + msz;
  float* Lm = Sm + msz;
  float* base = Lm + msz;
  float* u  = base + 0 * fld;
  float* v  = base + 1 * fld;
  float* dn = base + 2 * fld;
  float* nu = base + 3 * fld;
  float* nv = base + 4 * fld;
  float* nd = base + 5 * fld;
  float* dv = base + 6 * fld;   // divergence, later reused for q
  float* Yc = base + 7 * fld;   // stage temps (Zr reuses Yc, Zi reuses Ys)
  float* Ys = base + 8 * fld;
  float* Rh = base + 9 * fld;
  float* Ih = base + 10 * fld;
  if (ws_size < (3 * msz + 11 * fld) * sizeof(float)) return;

  const dim3 blk(256);
  const dim3 gAll((BATCH * N2) / 256);
  const dim3 gMat(N2 / 256);

  init_dft_mats<<<gMat, blk, 0, stream>>>(Cm, Sm, Lm);
  unpack_kernel<<<gAll, blk, 0, stream>>>(y0, u, v, dn);

  auto gemm = [&](const float* A, long long sA, const float* B, long long sB,
                  float* O, float alpha, float beta) {
    wmma_gemm256<<<dim3(4, 4, BATCH), dim3(128), 0, stream>>>(
        A, sA, B, sB, O, (long long)N2, alpha, beta);
  };

  const float invn2 = 1.0f / (float)N2;

  for (int t = 0; t < TSTEPS; ++t) {
    for (int it = 0; it < INNER_STEPS; ++it) {
      stencil_uv_kernel<<<gAll, blk, 0, stream>>>(u, v, nu, nv, dtp);
      divergence_kernel<<<gAll, blk, 0, stream>>>(nu, nv, dv);

      // Spectral Poisson solve: q = IFFT2( INV_LAM .* FFT2(div) ), all-real GEMM form.
      gemm(dv, N2, Cm, 0, Yc, 1.0f, 0.0f);          // Yc = div * C
      gemm(dv, N2, Sm, 0, Ys, 1.0f, 0.0f);          // Ys = div * S
      gemm(Cm, 0, Yc, N2, Rh, 1.0f, 0.0f);          // Rh = C*Yc
      gemm(Sm, 0, Ys, N2, Rh, -1.0f, 1.0f);         // Rh -= S*Ys          (= Re Dhat)
      gemm(Cm, 0, Ys, N2, Ih, 1.0f, 0.0f);          // Ih = C*Ys
      gemm(Sm, 0, Yc, N2, Ih, 1.0f, 1.0f);          // Ih += S*Yc          (= -Im Dhat)
      scale_eigs_kernel<<<gAll, blk, 0, stream>>>(Rh, Ih, Lm);
      gemm(Cm, 0, Rh, N2, Yc, 1.0f, 0.0f);          // Zr = C*Rh
      gemm(Sm, 0, Ih, N2, Yc, -1.0f, 1.0f);         // Zr -= S*Ih
      gemm(Cm, 0, Ih, N2, Ys, 1.0f, 0.0f);          // Zi = C*Ih
      gemm(Sm, 0, Rh, N2, Ys, 1.0f, 1.0f);          // Zi += S*Rh
      gemm(Yc, N2, Cm, 0, dv, invn2, 0.0f);         // q  = Zr*C / N^2
      gemm(Ys, N2, Sm, 0, dv, -invn2, 1.0f);        // q -= Zi*S / N^2

      apply_proj_kernel<<<gAll, blk, 0, stream>>>(nu, nv, dv);
      stencil_d_kernel<<<gAll, blk, 0, stream>>>(dn, u, v, nd, dtp);

      float* tp;
      tp = u;  u  = nu; nu = tp;
      tp = v;  v  = nv; nv = tp;
      tp = dn; dn = nd; nd = tp;
    }
    pack_kernel<<<gAll, blk, 0, stream>>>(u, v, dn, out, t);
  }
}